// MPC_32504312496442
// MI455X (gfx1250) — compile-verified
//
#include <hip/hip_runtime.h>

// MPC augmented-Lagrangian solver for MI455X (gfx1250).
// One wave32 per batch element; block-tridiagonal Newton solve with
// 16x16 blocks mapped onto V_WMMA_F32_16X16X4_F32.

constexpr int kNS = 12, kNC = 4, kT = 32, kNB = 64, kN = 16, kNZ = kT * kN;
constexpr float kDamp = 1e-4f;

typedef __attribute__((ext_vector_type(2))) float v2f;
typedef __attribute__((ext_vector_type(8))) float v8f;

// ---- WMMA fragment helpers (CDNA5 ISA 7.12.2 layouts, wave32) ----
// A 16x4 slice k of logical M(r,c)=base[r*rs+c*cs]: lane l holds row l&15,
// K = 4k + 2*(l>>4) + {0,1}.
__device__ __forceinline__ v2f frag_a(const float* m, int rs, int cs, int k, float alpha) {
  const int l = threadIdx.x;
  const int r = l & 15, h = l >> 4;
  const int kk = 4 * k + 2 * h;
  v2f a;
  a.x = alpha * m[r * rs + kk * cs];
  a.y = alpha * m[r * rs + (kk + 1) * cs];
  return a;
}
// B 4x16 slice k: lane l holds col l&15, K = 4k + 2*(l>>4) + {0,1}.
__device__ __forceinline__ v2f frag_b(const float* m, int rs, int cs, int k) {
  const int l = threadIdx.x;
  const int n = l & 15, h = l >> 4;
  const int kk = 4 * k + 2 * h;
  v2f b;
  b.x = m[kk * rs + n * cs];
  b.y = m[(kk + 1) * rs + n * cs];
  return b;
}
// C/D 16x16 f32: VGPR r, lane l -> row r + 8*(l>>4), col l&15.
__device__ __forceinline__ v8f frag_c_load(const float* m) {
  const int l = threadIdx.x;
  const int n = l & 15, h = l >> 4;
  v8f c;
#pragma unroll
  for (int r = 0; r < 8; ++r) c[r] = m[(r + 8 * h) * 16 + n];
  return c;
}
__device__ __forceinline__ void frag_c_store(float* m, v8f c) {
  const int l = threadIdx.x;
  const int n = l & 15, h = l >> 4;
#pragma unroll
  for (int r = 0; r < 8; ++r) m[(r + 8 * h) * 16 + n] = c[r];
}

// dst(16x16 row-major) = (alpha * X) * Y [+ Cin], X/Y via strides (transpose = swap).
__device__ __forceinline__ void mm16(float* dst, const float* X, int xr, int xc,
                                     const float* Y, int yr, int yc, float alpha,
                                     const float* Cin) {
  v8f acc = {};
  if (Cin) acc = frag_c_load(Cin);
#pragma unroll
  for (int k = 0; k < 4; ++k) {
    v2f a = frag_a(X, xr, xc, k, alpha);
    v2f b = frag_b(Y, yr, yc, k);
    // 8 args: (neg_a, A, neg_b, B, c_mod, C, reuse_a, reuse_b)
    acc = __builtin_amdgcn_wmma_f32_16x16x4_f32(false, a, false, b, (short)0, acc,
                                                false, false);
  }
  frag_c_store(dst, acc);
}

// Gauss-Jordan inverse of SPD 16x16 (lane = augmented column owner).
__device__ __forceinline__ void invert16(const float* S, float* out, float* Aug) {
  const int l = threadIdx.x;
#pragma unroll
  for (int i = 0; i < 16; ++i)
    Aug[i * 32 + l] = (l < 16) ? S[i * 16 + l] : ((l - 16 == i) ? 1.f : 0.f);
  __syncthreads();
  for (int k = 0; k < 16; ++k) {
    const float pr = 1.f / Aug[k * 32 + k];           // broadcast LDS read
    const float s = Aug[k * 32 + l] * pr;             // scaled pivot row (own col)
    Aug[k * 32 + l] = s;
#pragma unroll
    for (int i = 0; i < 16; ++i) {
      if (i == k) continue;
      const float f = Aug[i * 32 + k];                // pre-update multiplier
      Aug[i * 32 + l] -= f * s;
    }
    __builtin_amdgcn_wave_barrier();                  // keep LDS order across k
  }
  __syncthreads();
  if (l >= 16)
#pragma unroll
    for (int i = 0; i < 16; ++i) out[i * 16 + (l - 16)] = Aug[i * 32 + l];
  __syncthreads();
}

// D_t = diag(Q_t + damp + rho on x; rho*(mU+mL) on u) + [t<T-1] rho*CtC
__device__ __forceinline__ void buildD(int t, float rho, float* Dm, const float* Qv,
                                       const float* CTC, const float* muv,
                                       const float* mlv) {
  const int l = threadIdx.x;
  for (int i = l; i < 256; i += 32) {
    int r = i >> 4, cc = i & 15;
    float v = (t < kT - 1) ? rho * CTC[i] : 0.f;
    if (r == cc) {
      float q = Qv[t * 16 + r] + kDamp;
      q += (r < kNS) ? rho : rho * (muv[t * kNC + (r - kNS)] + mlv[t * kNC + (r - kNS)]);
      v += q;
    }
    Dm[i] = v;
  }
}

__global__ __launch_bounds__(32) void mpc_kernel(
    const float* __restrict__ x0, const float* __restrict__ Q,
    const float* __restrict__ c, const float* __restrict__ A,
    const float* __restrict__ B, const float* __restrict__ lo,
    const float* __restrict__ up, float* __restrict__ outX,
    float* __restrict__ outU) {
  __shared__ float CP[256], CTC[256], Tm[256], Dm[256], Aug[512];
  __shared__ float Sinv[kT * 256];                      // all Schur inverses (32 KB)
  __shared__ float zv[kNZ], Qv[kNZ], cv[kNZ], gv[kNZ], yv[kNZ], dzv[kNZ], rv[kNZ];
  __shared__ float lamE[kT * kNS];
  __shared__ float lamU[kT * kNC], lamL[kT * kNC], lov[kT * kNC], upv[kT * kNC];
  __shared__ float wuv[kT * kNC], wlv[kT * kNC], muv[kT * kNC], mlv[kT * kNC];
  __shared__ float x0s[16], tv[16];

  const int b = blockIdx.x;
  const int l = threadIdx.x;
  const int m = l & 15, h = l >> 4;

  __builtin_prefetch(Q + b * kNZ, 0, 3);                // global_prefetch_b8
  __builtin_prefetch(c + b * kNZ, 0, 3);

  // ---- stage inputs in LDS ----
  for (int i = l; i < kNZ; i += 32) {
    Qv[i] = Q[b * kNZ + i];
    cv[i] = c[b * kNZ + i];
  }
  for (int i = l; i < kT * kNC; i += 32) {
    lov[i] = lo[b * kT * kNC + i];
    upv[i] = up[b * kT * kNC + i];
    lamU[i] = 0.f;
    lamL[i] = 0.f;
  }
  for (int i = l; i < kT * kNS; i += 32) lamE[i] = 0.f;
  if (l < 16) x0s[l] = (l < kNS) ? x0[b * kNS + l] : 0.f;
  for (int i = l; i < 256; i += 32) {                   // CP = [A B] padded to 16x16
    int r = i >> 4, cc = i & 15;
    float v = 0.f;
    if (r < kNS) v = (cc < kNS) ? A[r * kNS + cc] : B[r * kNC + (cc - kNS)];
    CP[i] = v;
  }
  __syncthreads();

  mm16(CTC, CP, 1, 16, CP, 16, 1, 1.0f, nullptr);       // CTC = CPt * CP (WMMA)
  __syncthreads();

  // ---- z init: x_t = A^t x0, u = 0 ----
  if (l < 16) zv[l] = (l < kNS) ? x0s[l] : 0.f;
  __syncthreads();
  for (int t = 1; t < kT; ++t) {
    if (l < 16) {
      float v = 0.f;
      if (l < kNS) {
        float s = 0.f;
        for (int j = 0; j < 16; ++j) s += CP[l * 16 + j] * zv[(t - 1) * 16 + j];
        v = s;
      }
      zv[t * 16 + l] = v;
    }
    __syncthreads();
  }

  float rho = 1.0f;
  for (int al = 0; al < 3; ++al) {
    for (int it = 0; it < 3; ++it) {
      // ---- v = rho*r + lamE (x rows, padded); bound slacks & active masks ----
      for (int t = h; t < kT; t += 2) {
        float r_ = 0.f;
        if (m < kNS) {
          if (t == 0)
            r_ = zv[m] - x0s[m];
          else {
            float s = 0.f;
            for (int j = 0; j < 16; ++j) s += CP[m * 16 + j] * zv[(t - 1) * 16 + j];
            r_ = zv[t * 16 + m] - s;
          }
        }
        rv[t * 16 + m] = rho * r_ + ((m < kNS) ? lamE[t * kNS + m] : 0.f);
      }
      for (int i = l; i < kT * kNC; i += 32) {
        int t = i >> 2, j = i & 3;
        float u = zv[t * 16 + kNS + j];
        float au = rho * (u - upv[i]) + lamU[i];
        float ad = rho * (lov[i] - u) + lamL[i];
        wuv[i] = au > 0.f ? au : 0.f;
        wlv[i] = ad > 0.f ? ad : 0.f;
        muv[i] = au > 0.f ? 1.f : 0.f;
        mlv[i] = ad > 0.f ? 1.f : 0.f;
      }
      __syncthreads();
      // ---- gradient g = Q.z + c + Jt v + Gt w ----
      for (int t = h; t < kT; t += 2) {
        float g = Qv[t * 16 + m] * zv[t * 16 + m] + cv[t * 16 + m];
        if (m < kNS) {
          g += rv[t * 16 + m];
          if (t < kT - 1) {
            float s = 0.f;
            for (int r = 0; r < kNS; ++r) s += CP[r * 16 + m] * rv[(t + 1) * 16 + r];
            g -= s;
          }
        } else {
          g += wuv[t * kNC + (m - kNS)] - wlv[t * kNC + (m - kNS)];
        }
        gv[t * 16 + m] = g;
      }
      __syncthreads();
      // ---- block-Thomas forward sweep: S_t = D_t - rho^2 CP Sinv_{t-1} CPt ----
      buildD(0, rho, Dm, Qv, CTC, muv, mlv);
      __syncthreads();
      invert16(Dm, &Sinv[0], Aug);
      if (l < 16) yv[l] = gv[l];
      __syncthreads();
      for (int t = 1; t < kT; ++t) {
        mm16(Tm, &Sinv[(t - 1) * 256], 16, 1, CP, 1, 16, 1.0f, nullptr);  // Sinv*CPt
        __syncthreads();
        buildD(t, rho, Dm, Qv, CTC, muv, mlv);
        __syncthreads();
        mm16(Dm, CP, 16, 1, Tm, 16, 1, -rho * rho, Dm);  // Dm += (-rho^2 CP)*Tm
        __syncthreads();
        if (l < 16) {                                    // tv = Sinv_{t-1} y_{t-1}
          float s = 0.f;
          for (int j = 0; j < 16; ++j)
            s += Sinv[(t - 1) * 256 + l * 16 + j] * yv[(t - 1) * 16 + j];
          tv[l] = s;
        }
        __syncthreads();
        if (l < 16) {                                    // y_t = g_t + rho*CP*tv
          float yy = gv[t * 16 + l];
          if (l < kNS) {
            float s = 0.f;
            for (int j = 0; j < 16; ++j) s += CP[l * 16 + j] * tv[j];
            yy += rho * s;
          }
          yv[t * 16 + l] = yy;
        }
        __syncthreads();
        invert16(Dm, &Sinv[t * 256], Aug);
        __syncthreads();
      }
      // ---- back substitution ----
      if (l < 16) {
        float s = 0.f;
        for (int j = 0; j < 16; ++j)
          s += Sinv[(kT - 1) * 256 + l * 16 + j] * yv[(kT - 1) * 16 + j];
        dzv[(kT - 1) * 16 + l] = s;
      }
      __syncthreads();
      for (int t = kT - 2; t >= 0; --t) {
        if (l < 16) {                                    // tv = y_t + rho*CPt*dz_{t+1}
          float s = 0.f;
          for (int r = 0; r < kNS; ++r) s += CP[r * 16 + l] * dzv[(t + 1) * 16 + r];
          tv[l] = yv[t * 16 + l] + rho * s;
        }
        __syncthreads();
        if (l < 16) {
          float s = 0.f;
          for (int j = 0; j < 16; ++j) s += Sinv[t * 256 + l * 16 + j] * tv[j];
          dzv[t * 16 + l] = s;
        }
        __syncthreads();
      }
      for (int i = l; i < kNZ; i += 32) zv[i] -= dzv[i];
      __syncthreads();
    }
    // ---- AL multiplier update ----
    for (int t = h; t < kT; t += 2) {
      if (m < kNS) {
        float r_;
        if (t == 0)
          r_ = zv[m] - x0s[m];
        else {
          float s = 0.f;
          for (int j = 0; j < 16; ++j) s += CP[m * 16 + j] * zv[(t - 1) * 16 + j];
          r_ = zv[t * 16 + m] - s;
        }
        lamE[t * kNS + m] += rho * r_;
      }
    }
    for (int i = l; i < kT * kNC; i += 32) {
      int t = i >> 2, j = i & 3;
      float u = zv[t * 16 + kNS + j];
      float nu = lamU[i] + rho * (u - upv[i]);
      float nl = lamL[i] + rho * (lov[i] - u);
      lamU[i] = nu > 0.f ? nu : 0.f;
      lamL[i] = nl > 0.f ? nl : 0.f;
    }
    __syncthreads();
    rho = rho * 10.f;
    if (rho > 100.f) rho = 100.f;
  }

  // ---- outputs: x (NB,T,NS) then u (NB,T,NC), flat ----
  for (int i = l; i < kT * kNS; i += 32) {
    int t = i / kNS, mm = i - t * kNS;
    outX[b * kT * kNS + i] = zv[t * 16 + mm];
  }
  for (int i = l; i < kT * kNC; i += 32) {
    int t = i >> 2, j = i & 3;
    outU[b * kT * kNC + i] = zv[t * 16 + kNS + j];
  }
}

extern "C" void kernel_launch(void* const* d_in, const int* in_sizes, int n_in,
                              void* d_out, int out_size, void* d_ws, size_t ws_size,
                              hipStream_t stream) {
  (void)in_sizes; (void)n_in; (void)d_ws; (void)ws_size; (void)out_size;
  const float* x0 = (const float*)d_in[0];
  const float* Q  = (const float*)d_in[1];
  const float* c  = (const float*)d_in[2];
  const float* A  = (const float*)d_in[3];
  const float* B  = (const float*)d_in[4];
  const float* lo = (const float*)d_in[5];
  const float* up = (const float*)d_in[6];
  float* out = (float*)d_out;
  float* outX = out;
  float* outU = out + (size_t)kNB * kT * kNS;
  mpc_kernel<<<dim3(kNB), dim3(32), 0, stream>>>(x0, Q, c, A, B, lo, up, outX, outU);
}